// HGNN_8993661517924
// MI455X (gfx1250) — compile-verified
//
#include <hip/hip_runtime.h>
#include <math.h>

#define NN 100000
#define NE 800000
#define NEG_SLOPE 0.2f

typedef float v2f __attribute__((ext_vector_type(2)));
typedef float v8f __attribute__((ext_vector_type(8)));
typedef unsigned int u32x4 __attribute__((ext_vector_type(4)));
typedef int i32x4 __attribute__((ext_vector_type(4)));
typedef int i32x8 __attribute__((ext_vector_type(8)));

#if defined(__has_builtin)
#if __has_builtin(__builtin_amdgcn_tensor_load_to_lds) && \
    __has_builtin(__builtin_amdgcn_s_wait_tensorcnt)
#define HAVE_TDM_BUILTIN 1
#endif
#endif
#ifndef HAVE_TDM_BUILTIN
#define HAVE_TDM_BUILTIN 0
#endif

#define XSTR2(x) #x
#define XSTR(x) XSTR2(x)
#pragma message("CDNA5-PROBE clang=" XSTR(__clang_major__) " HAVE_TDM_BUILTIN=" XSTR(HAVE_TDM_BUILTIN))

// ---------------- TDM: DMA a 16x64 fp32 tile (row stride 64) into LDS -------
// LDS layout gets 4 DWORDs of padding every 64 DWORDs -> row stride 68 floats,
// matching the bank-conflict-free layout the WMMA A-fragment reads use.
__device__ __forceinline__ void tdm_load_tile_16x64(const float* __restrict__ gsrc,
                                                    void* ldsDst) {
  unsigned long long ga = (unsigned long long)(const void*)gsrc;
  unsigned int la = (unsigned int)(unsigned long long)ldsDst;  // low 32b = LDS offset
  u32x4 g0;
  g0[0] = 1u;                                            // count=1, user descriptor
  g0[1] = la;                                            // lds_addr
  g0[2] = (unsigned int)ga;                              // global_addr[31:0]
  g0[3] = (unsigned int)((ga >> 32) & 0x01FFFFFFu) | (2u << 30);  // addr hi | type=2
  i32x8 g1;
  // data_size=4B(2), pad_enable, pad_interval=5 (64 DW), pad_amount=3 (4 DW)
  g1[0] = (2 << 16) | (1 << 20) | (5 << 22) | (3 << 25);
  g1[1] = (64 << 16);        // tensor_dim0 = 64 (bits 63:48 of group)
  g1[2] = (16 << 16);        // tensor_dim0 hi=0 | tensor_dim1 lo = 16
  g1[3] = (64 << 16);        // tensor_dim1 hi=0 | tile_dim0 = 64
  g1[4] = 16;                // tile_dim1 = 16, tile_dim2 = 0
  g1[5] = 64;                // tensor_dim0_stride = 64 elements
  g1[6] = 0;
  g1[7] = 0;
#if HAVE_TDM_BUILTIN
  i32x4 z4 = {0, 0, 0, 0};
#if defined(__clang_major__) && __clang_major__ >= 23
  i32x8 z8 = {0, 0, 0, 0, 0, 0, 0, 0};
  __builtin_amdgcn_tensor_load_to_lds(g0, g1, z4, z4, z8, 0);
#else
  __builtin_amdgcn_tensor_load_to_lds(g0, g1, z4, z4, 0);
#endif
  __builtin_amdgcn_s_wait_tensorcnt(0);
#else
  // 2-descriptor-group form: groups 2/3 are NULL (tensor is 2-D).
  asm volatile("tensor_load_to_lds %0, %1\n\t"
               "s_wait_tensorcnt 0x0"
               :
               : "s"(g0), "s"(g1)
               : "memory");
#endif
}

// Stage a 16x64 tile (+ stride-68 padding) into LDS; whole block participates.
__device__ __forceinline__ void stage_tile_16x64(const float* __restrict__ gsrc,
                                                 float* As, int nthreads) {
  if (threadIdx.x < 32) {               // one full-EXEC wave issues the DMA
    tdm_load_tile_16x64(gsrc, As);
  }
  __syncthreads();
  (void)nthreads;
}

// ---------------- WMMA fp32 16x16x4 tile helpers ----------------
// A fragment: lane<16 -> row=lane&15, K=kk,kk+1 ; lane>=16 -> K=kk+2,kk+3
// B fragment: col = lane&15, same K split.  C: reg r -> row r (+8 for lanes>=16).
__device__ __forceinline__ v8f wmma_k_loop(const float* __restrict__ As, int lda,
                                           const float* __restrict__ B, int ldb,
                                           int kLen, v8f c) {
  const int lane = threadIdx.x & 31;
  const int l15  = lane & 15;
  const int kOff = (lane & 16) ? 2 : 0;
  for (int kk = 0; kk < kLen; kk += 4) {
    const int k0 = kk + kOff;
    v2f a, b;
    a.x = As[l15 * lda + k0];
    a.y = As[l15 * lda + k0 + 1];
    b.x = B[(long)k0 * ldb + l15];
    b.y = B[(long)(k0 + 1) * ldb + l15];
    c = __builtin_amdgcn_wmma_f32_16x16x4_f32(false, a, false, b, (short)0, c,
                                              false, false);
  }
  return c;
}

__device__ __forceinline__ void store_tile(float* __restrict__ C, long ldc,
                                           long row0, int col0, v8f c,
                                           const float* __restrict__ bias) {
  const int lane = threadIdx.x & 31;
  const int l15  = lane & 15;
  const int r0   = (lane & 16) ? 8 : 0;
  const float bv = bias ? bias[col0 + l15] : 0.0f;
#pragma unroll
  for (int r = 0; r < 8; ++r)
    C[(row0 + r0 + r) * ldc + col0 + l15] = c[r] + bv;
}

// ---------------- GEMM 1: h[N,256] = x[N,64] @ W_att(hio -> [64, H*64]) ------
__global__ void __launch_bounds__(256)
gemm_att_kernel(const float* __restrict__ x, const float* __restrict__ W_att,
                float* __restrict__ h) {
  __shared__ __align__(16) float As[16 * 68];
  const long row0 = (long)blockIdx.x * 16;
  stage_tile_16x64(x + row0 * 64, As, 256);
  const int wave = threadIdx.x >> 5;
#pragma unroll
  for (int t = 0; t < 2; ++t) {
    const int ct   = wave * 2 + t;   // 16 column tiles of 16 -> 256 cols
    const int col0 = ct * 16;
    const int head = col0 >> 6;      // tile never crosses a head boundary
    const int o0   = col0 & 63;
    v8f c = {};
    // B element [k, head*64+o] = W_att[head*4096 + k*64 + o]
    c = wmma_k_loop(As, 68, W_att + head * 4096 + o0, 64, 64, c);
    store_tile(h, 256, row0, col0, c, nullptr);
  }
}

// ---------------- per-node attention logits ---------------------------------
__global__ void __launch_bounds__(256)
node_scores_kernel(const float* __restrict__ h, const float* __restrict__ a_src,
                   const float* __restrict__ a_dst, float* __restrict__ s_src,
                   float* __restrict__ s_dst) {
  const long n  = ((long)blockIdx.x * blockDim.x + threadIdx.x) >> 5;
  const int lane = threadIdx.x & 31;
  if (n >= NN) return;
  const float* hn = h + n * 256;
#pragma unroll
  for (int hd = 0; hd < 4; ++hd) {
    float v0 = hn[hd * 64 + lane], v1 = hn[hd * 64 + lane + 32];
    float ss = v0 * a_src[hd * 64 + lane] + v1 * a_src[hd * 64 + lane + 32];
    float sd = v0 * a_dst[hd * 64 + lane] + v1 * a_dst[hd * 64 + lane + 32];
    for (int msk = 16; msk; msk >>= 1) {
      ss += __shfl_xor(ss, msk, 32);
      sd += __shfl_xor(sd, msk, 32);
    }
    if (lane == 0) { s_src[n * 4 + hd] = ss; s_dst[n * 4 + hd] = sd; }
  }
}

// ---------------- misc ------------------------------------------------------
__global__ void fill_kernel(float* __restrict__ p, float v, long n) {
  long i = (long)blockIdx.x * blockDim.x + threadIdx.x;
  const long stride = (long)gridDim.x * blockDim.x;
  for (; i < n; i += stride) p[i] = v;
}

__device__ __forceinline__ void atomicMaxF(float* addr, float v) {
  if (v >= 0.0f) atomicMax((int*)addr, __float_as_int(v));
  else atomicMin((unsigned int*)addr, (unsigned int)__float_as_int(v));
}

// scores + leaky-relu + segment-max (atomic) + degree count, one edge pass
__global__ void edge_score_max_kernel(const int* __restrict__ src,
                                      const int* __restrict__ dst,
                                      const float* __restrict__ s_src,
                                      const float* __restrict__ s_dst,
                                      float* __restrict__ score,
                                      float* __restrict__ m,
                                      float* __restrict__ deg) {
  int e = blockIdx.x * blockDim.x + threadIdx.x;
  if (e >= NE) return;
  int s = src[e], d = dst[e];
  atomicAdd(&deg[d], 1.0f);
#pragma unroll
  for (int hd = 0; hd < 4; ++hd) {
    float sc = s_src[(long)s * 4 + hd] + s_dst[(long)d * 4 + hd];
    sc = (sc > 0.0f) ? sc : NEG_SLOPE * sc;
    score[(long)e * 4 + hd] = sc;
    atomicMaxF(&m[(long)d * 4 + hd], sc);
  }
}

__global__ void edge_exp_kernel(const int* __restrict__ dst,
                                float* __restrict__ score,
                                const float* __restrict__ m,
                                float* __restrict__ den) {
  int e = blockIdx.x * blockDim.x + threadIdx.x;
  if (e >= NE) return;
  int d = dst[e];
#pragma unroll
  for (int hd = 0; hd < 4; ++hd) {
    float ex = __expf(score[(long)e * 4 + hd] - m[(long)d * 4 + hd]);
    score[(long)e * 4 + hd] = ex;
    atomicAdd(&den[(long)d * 4 + hd], ex);
  }
}

// alpha * h[src] scatter, fused head-mean (x0.25) -> local[N,64]
__global__ void __launch_bounds__(256)
edge_agg_kernel(const int* __restrict__ src, const int* __restrict__ dst,
                const float* __restrict__ ex, const float* __restrict__ den,
                const float* __restrict__ h, float* __restrict__ local) {
  const long e = ((long)blockIdx.x * blockDim.x + threadIdx.x) >> 5;
  const int lane = threadIdx.x & 31;
  if (e >= NE) return;
  const int s = src[e], d = dst[e];
  // lanes 0..3 compute alpha_h, then broadcast (avoids 32x redundant loads)
  float aval = 0.0f;
  if (lane < 4)
    aval = 0.25f * ex[e * 4 + lane] / fmaxf(den[(long)d * 4 + lane], 1e-16f);
  float acc0 = 0.f, acc1 = 0.f;
#pragma unroll
  for (int hd = 0; hd < 4; ++hd) {
    float a = __shfl(aval, hd, 32);
    acc0 += a * h[(long)s * 256 + hd * 64 + lane];
    acc1 += a * h[(long)s * 256 + hd * 64 + lane + 32];
  }
  atomicAdd(&local[(long)d * 64 + lane], acc0);
  atomicAdd(&local[(long)d * 64 + lane + 32], acc1);
}

// hop 1: acc1[d] += x[s]
__global__ void __launch_bounds__(256)
hop_gather_kernel(const int* __restrict__ src, const int* __restrict__ dst,
                  const float* __restrict__ xin, float* __restrict__ acc) {
  const long e = ((long)blockIdx.x * blockDim.x + threadIdx.x) >> 5;
  const int lane = threadIdx.x & 31;
  if (e >= NE) return;
  const int s = src[e], d = dst[e];
  atomicAdd(&acc[(long)d * 64 + lane], xin[(long)s * 64 + lane]);
  atomicAdd(&acc[(long)d * 64 + lane + 32], xin[(long)s * 64 + lane + 32]);
}

// hop 2 with hop-1's mean folded in: acc2[d] += acc1[s] / max(deg[s],1)
__global__ void __launch_bounds__(256)
hop_gather_scaled_kernel(const int* __restrict__ src, const int* __restrict__ dst,
                         const float* __restrict__ xin,
                         const float* __restrict__ deg, float* __restrict__ acc) {
  const long e = ((long)blockIdx.x * blockDim.x + threadIdx.x) >> 5;
  const int lane = threadIdx.x & 31;
  if (e >= NE) return;
  const int s = src[e], d = dst[e];
  const float inv = 1.0f / fmaxf(deg[s], 1.0f);
  atomicAdd(&acc[(long)d * 64 + lane], inv * xin[(long)s * 64 + lane]);
  atomicAdd(&acc[(long)d * 64 + lane + 32], inv * xin[(long)s * 64 + lane + 32]);
}

// ---------------- GEMM 2: glob_out = (acc2/deg) @ W_g + b_g -----------------
__global__ void __launch_bounds__(128)
gemm_glob_kernel(const float* __restrict__ g, const float* __restrict__ deg,
                 const float* __restrict__ Wg, const float* __restrict__ bg,
                 float* __restrict__ out) {
  __shared__ __align__(16) float As[16 * 68];
  const long row0 = (long)blockIdx.x * 16;
  // manual staging: divide by degree (hop-2 mean) while loading
  for (int i = threadIdx.x; i < 16 * 64; i += 128) {
    int r = i >> 6, k = i & 63;
    As[r * 68 + k] = g[(row0 + r) * 64 + k] / fmaxf(deg[row0 + r], 1.0f);
  }
  __syncthreads();
  const int col0 = (threadIdx.x >> 5) * 16;
  v8f c = {};
  c = wmma_k_loop(As, 68, Wg + col0, 64, 64, c);
  store_tile(out, 64, row0, col0, c, bg);
}

// ---------------- GEMM 3: out = local@Ww[:64] + glob@Ww[64:] + b_w ----------
__global__ void __launch_bounds__(128)
gemm_out_kernel(const float* __restrict__ local, const float* __restrict__ glob,
                const float* __restrict__ Ww, const float* __restrict__ bw,
                float* __restrict__ out) {
  __shared__ __align__(16) float As[16 * 68];
  const long row0 = (long)blockIdx.x * 16;
  const int col0 = (threadIdx.x >> 5) * 16;
  v8f c = {};
  stage_tile_16x64(local + row0 * 64, As, 128);
  c = wmma_k_loop(As, 68, Ww + col0, 64, 64, c);
  __syncthreads();   // everyone done reading LDS before overwrite
  stage_tile_16x64(glob + row0 * 64, As, 128);
  c = wmma_k_loop(As, 68, Ww + 64 * 64 + col0, 64, 64, c);
  store_tile(out, 64, row0, col0, c, bw);
}

// ---------------- launcher --------------------------------------------------
extern "C" void kernel_launch(void* const* d_in, const int* in_sizes, int n_in,
                              void* d_out, int out_size, void* d_ws, size_t ws_size,
                              hipStream_t stream) {
  const float* x     = (const float*)d_in[0];
  const int*   ei    = (const int*)d_in[1];
  const float* W_att = (const float*)d_in[2];
  const float* a_src = (const float*)d_in[3];
  const float* a_dst = (const float*)d_in[4];
  const float* W_g   = (const float*)d_in[5];
  const float* b_g   = (const float*)d_in[6];
  const float* W_w   = (const float*)d_in[7];
  const float* b_w   = (const float*)d_in[8];
  float* out = (float*)d_out;
  const int* src = ei;
  const int* dst = ei + NE;
  (void)in_sizes; (void)n_in; (void)out_size; (void)ws_size;

  char* ws = (char*)d_ws;
  size_t off = 0;
  auto alloc = [&](size_t bytes) -> float* {
    float* p = (float*)(ws + off);
    off += (bytes + 255) & ~(size_t)255;
    return p;
  };
  // zero-initialized accumulators first, contiguous -> single fill launch
  float* den   = alloc((size_t)NN * 4 * 4);
  float* deg   = alloc((size_t)NN * 4);
  float* local = alloc((size_t)NN * 64 * 4);
  float* acc1  = alloc((size_t)NN * 64 * 4);
  float* acc2  = alloc((size_t)NN * 64 * 4);
  const long zero_len = ((char*)(acc2 + (size_t)NN * 64) - (char*)den) / 4;
  // rest
  float* m     = alloc((size_t)NN * 4 * 4);
  float* h     = alloc((size_t)NN * 256 * 4);  // [N, H*64]
  float* ssrc  = alloc((size_t)NN * 4 * 4);
  float* sdst  = alloc((size_t)NN * 4 * 4);
  float* score = alloc((size_t)NE * 4 * 4);    // reused as exp()
  float* gout  = alloc((size_t)NN * 64 * 4);

  fill_kernel<<<4096, 256, 0, stream>>>(den, 0.f, zero_len);
  fill_kernel<<<1024, 256, 0, stream>>>(m, -INFINITY, (long)NN * 4);

  gemm_att_kernel<<<NN / 16, 256, 0, stream>>>(x, W_att, h);
  node_scores_kernel<<<(NN + 7) / 8, 256, 0, stream>>>(h, a_src, a_dst, ssrc, sdst);
  edge_score_max_kernel<<<(NE + 255) / 256, 256, 0, stream>>>(src, dst, ssrc, sdst,
                                                              score, m, deg);
  edge_exp_kernel<<<(NE + 255) / 256, 256, 0, stream>>>(dst, score, m, den);
  edge_agg_kernel<<<(NE + 7) / 8, 256, 0, stream>>>(src, dst, score, den, h, local);
  hop_gather_kernel<<<(NE + 7) / 8, 256, 0, stream>>>(src, dst, x, acc1);
  hop_gather_scaled_kernel<<<(NE + 7) / 8, 256, 0, stream>>>(src, dst, acc1, deg, acc2);
  gemm_glob_kernel<<<NN / 16, 128, 0, stream>>>(acc2, deg, W_g, b_g, gout);
  gemm_out_kernel<<<NN / 16, 128, 0, stream>>>(local, gout, W_w, b_w, out);
}